// MGN_NET_77635828842663
// MI455X (gfx1250) — compile-verified
//
#include <hip/hip_runtime.h>
#include <hip/hip_bf16.h>

// CDNA5 / gfx1250 — wave32, WMMA f32 16x16x4.
typedef __attribute__((ext_vector_type(2))) float v2f;
typedef __attribute__((ext_vector_type(4))) float v4f;
typedef __attribute__((ext_vector_type(8))) float v8f;

#define WAVES 8   // 8 waves * 16 edges = 128 edges per block

// ---------------------------------------------------------------------------
// Degree (scatter count of dst)
// ---------------------------------------------------------------------------
__global__ void deg_kernel(const int* __restrict__ ei, float* __restrict__ deg, int E) {
    int e = blockIdx.x * blockDim.x + threadIdx.x;
    if (e < E) atomicAdd(&deg[ei[E + e]], 1.0f);
}

// ---------------------------------------------------------------------------
// Fused NNConv edge pass (templated on layer shape so all index math and
// trip counts are compile-time):
//   ew = relu(edge_attr @ w_nn + b_nn)    (WMMA f32 16x16x4, K=6 padded to 8)
//   msg[e,o] = sum_i x[src[e],i] * ew[e, i*DOUT+o]   (LDS atomics)
//   agg[dst[e],o] += msg[e,o]                        (global atomics)
// One wave owns a 16-edge tile; loops over 16-wide column tiles of DIN*DOUT.
// All lane-dependent selects are value-selects (v_cndmask), never predicated
// loads, so EXEC stays all-ones through the WMMA issue points.
// ---------------------------------------------------------------------------
template<int DIN, int DOUT>
__global__ __launch_bounds__(256) void nnconv_edge_kernel(
    const float* __restrict__ x,      // [N, DIN]
    const float* __restrict__ attr,   // [E, 6]
    const int*   __restrict__ ei,     // [2, E]
    const float* __restrict__ w_nn,   // [6, DIN*DOUT]
    const float* __restrict__ b_nn,   // [DIN*DOUT]
    float*       __restrict__ agg,    // [N, DOUT]
    int E)
{
    constexpr int DD  = DIN * DOUT;
    constexpr int NCT = (DD + 15) >> 4;

    extern __shared__ float lds[];
    float* sW       = lds;                       // 6*DD
    float* sBias    = sW + 6 * DD;               // DD
    float* sXbase   = sBias + DD;                // WAVES*16*DIN
    float* sMsgBase = sXbase + WAVES * 16 * DIN; // WAVES*16*DOUT

    const int tid  = threadIdx.x;
    const int lane = tid & 31;
    const int wave = tid >> 5;

    // cooperative preload of weights + bias into LDS
    for (int t = tid; t < 6 * DD; t += 256) sW[t] = w_nn[t];
    for (int t = tid; t < DD;     t += 256) sBias[t] = b_nn[t];
    __syncthreads();

    const int e0 = (blockIdx.x * WAVES + wave) * 16;
    const int* srcArr = ei;
    const int* dstArr = ei + E;

    float* xs  = sXbase   + wave * 16 * DIN;
    float* msg = sMsgBase + wave * 16 * DOUT;

    // zero message accumulator
    for (int t = lane; t < 16 * DOUT; t += 32) msg[t] = 0.0f;

    // gather x[src] for the 16 edges of this wave
    for (int t = lane; t < 16 * DIN; t += 32) {
        int el = t / DIN;
        int i  = t - el * DIN;
        int e  = e0 + el; if (e >= E) e = E - 1;
        xs[t] = x[srcArr[e] * DIN + i];
    }

    // A fragment: 16x4 f32 layout (lanes 0-15: K=0,1 ; lanes 16-31: K=2,3)
    const int m  = lane & 15;
    const int hi = lane >> 4;
    const int kb = hi * 2;
    int ea = e0 + m; if (ea >= E) ea = E - 1;
    const float* arow = attr + (size_t)ea * 6;
    // unconditional loads (addresses valid in all lanes), lane-select by value
    const float a4 = arow[4], a5 = arow[5];
    v2f a1, a2;
    a1.x = arow[kb];
    a1.y = arow[kb + 1];
    a2.x = hi ? 0.0f : a4;        // second K-step covers K=4..7 (6,7 are pad)
    a2.y = hi ? 0.0f : a5;

    for (int ct = 0; ct < NCT; ++ct) {
        const int  col = ct * 16 + m;          // lane's output column (N index)
        const bool cv  = (DD % 16 == 0) || (col < DD);
        const int  cc  = cv ? col : 0;
        // B fragment: 4x16 f32 (lanes 0-15: K=0,1 ; lanes 16-31: K=2,3).
        // Load all four rows unconditionally, then value-select (no exec mask).
        const float w01 = sW[(kb    ) * DD + cc];
        const float w23 = sW[(kb + 1) * DD + cc];
        const float w4  = sW[4 * DD + cc];
        const float w5  = sW[5 * DD + cc];
        v2f b1, b2;
        b1.x = cv ? w01 : 0.0f;
        b1.y = cv ? w23 : 0.0f;
        b2.x = (cv && !hi) ? w4 : 0.0f;
        b2.y = (cv && !hi) ? w5 : 0.0f;

        v8f c = {};
        c = __builtin_amdgcn_wmma_f32_16x16x4_f32(false, a1, false, b1, (short)0, c, false, false);
        c = __builtin_amdgcn_wmma_f32_16x16x4_f32(false, a2, false, b2, (short)0, c, false, false);

        if (cv) {
            const float bias = sBias[col];
            const int i = col / DOUT;            // compile-time-constant divisor
            const int o = col - i * DOUT;
            // batch the 8 xs loads under one DS wait, then fire 8 ds_add_f32
            float xv[8];
#pragma unroll
            for (int r = 0; r < 8; ++r)
                xv[r] = xs[(r + hi * 8) * DIN + i];
#pragma unroll
            for (int r = 0; r < 8; ++r) {
                float ew = c[r] + bias;
                ew = ew > 0.0f ? ew : 0.0f;          // ReLU
                atomicAdd(&msg[(r + hi * 8) * DOUT + o], ew * xv[r]);
            }
        }
    }
    __syncthreads();   // order LDS atomics before readback

    // scatter messages to agg (mean divides later in node kernel)
    for (int t = lane; t < 16 * DOUT; t += 32) {
        int el = t / DOUT;
        int o  = t - el * DOUT;
        int e  = e0 + el;
        if (e < E) atomicAdd(&agg[(size_t)dstArr[e] * DOUT + o], msg[t]);
    }
}

// ---------------------------------------------------------------------------
// Node update: x_out = relu(agg/max(deg,1) + x @ lin^T + bias)
// ---------------------------------------------------------------------------
template<int DIN, int DOUT>
__global__ void node_update_kernel(
    const float* __restrict__ agg, const float* __restrict__ deg,
    const float* __restrict__ xin, const float* __restrict__ lin,
    const float* __restrict__ bias, float* __restrict__ xout, int Nn)
{
    int idx = blockIdx.x * blockDim.x + threadIdx.x;
    if (idx >= Nn * DOUT) return;
    int n = idx / DOUT;
    int o = idx - n * DOUT;
    float d = deg[n]; d = d > 1.0f ? d : 1.0f;
    float v = agg[idx] / d;
    const float* xr = xin + (size_t)n * DIN;
    const float* lr = lin + (size_t)o * DIN;
    float s = 0.0f;
#pragma unroll
    for (int i = 0; i < DIN; ++i) s += xr[i] * lr[i];
    v += s + bias[o];
    xout[idx] = v > 0.0f ? v : 0.0f;
}

// ---------------------------------------------------------------------------
// Pairwise L1 distance: out[i*N+j] = sum_f |x3[j,f] - x3[i,f]|  (F = 5)
// Each thread emits one v4f of the 67MB output; NT store (write-once,
// never re-read) keeps the 192MB L2 hot for everything else.
// ---------------------------------------------------------------------------
__global__ __launch_bounds__(256) void cbt_kernel(
    const float* __restrict__ x3, float* __restrict__ out, int Nn)
{
    const int i  = blockIdx.y;
    const int j0 = (blockIdx.x * blockDim.x + threadIdx.x) * 4;
    if (j0 >= Nn) return;

    const float xi0 = x3[i*5+0], xi1 = x3[i*5+1], xi2 = x3[i*5+2],
                xi3 = x3[i*5+3], xi4 = x3[i*5+4];

    // rows j0..j0+3 are exactly 20 consecutive floats = 5 aligned 16B vectors
    const v4f* p = (const v4f*)(x3 + (size_t)j0 * 5);
    v4f r0 = p[0], r1 = p[1], r2 = p[2], r3 = p[3], r4 = p[4];

    v4f res;
    res.x = fabsf(r0.x-xi0)+fabsf(r0.y-xi1)+fabsf(r0.z-xi2)+fabsf(r0.w-xi3)+fabsf(r1.x-xi4);
    res.y = fabsf(r1.y-xi0)+fabsf(r1.z-xi1)+fabsf(r1.w-xi2)+fabsf(r2.x-xi3)+fabsf(r2.y-xi4);
    res.z = fabsf(r2.z-xi0)+fabsf(r2.w-xi1)+fabsf(r3.x-xi2)+fabsf(r3.y-xi3)+fabsf(r3.z-xi4);
    res.w = fabsf(r3.w-xi0)+fabsf(r4.x-xi1)+fabsf(r4.y-xi2)+fabsf(r4.z-xi3)+fabsf(r4.w-xi4);

    __builtin_nontemporal_store(res, (v4f*)(out + (size_t)i * Nn + j0));
}

// ---------------------------------------------------------------------------
extern "C" void kernel_launch(void* const* d_in, const int* in_sizes, int n_in,
                              void* d_out, int out_size, void* d_ws, size_t ws_size,
                              hipStream_t stream) {
    const int Nn = 4096, E = 131072;

    const float* x     = (const float*)d_in[0];
    const float* eattr = (const float*)d_in[1];
    const int*   ei    = (const int*)  d_in[2];

    const float* w_nn[3] = { (const float*)d_in[3],  (const float*)d_in[7],  (const float*)d_in[11] };
    const float* b_nn[3] = { (const float*)d_in[4],  (const float*)d_in[8],  (const float*)d_in[12] };
    const float* lin [3] = { (const float*)d_in[5],  (const float*)d_in[9],  (const float*)d_in[13] };
    const float* bias[3] = { (const float*)d_in[6],  (const float*)d_in[10], (const float*)d_in[14] };

    // workspace: deg[N] | agg[N*36] | bufA[N*36] | bufB[N*36]
    float* deg  = (float*)d_ws;
    float* agg  = deg  + Nn;
    float* bufA = agg  + (size_t)Nn * 36;
    float* bufB = bufA + (size_t)Nn * 36;

    (void)hipMemsetAsync(deg, 0, Nn * sizeof(float), stream);
    deg_kernel<<<E / 256, 256, 0, stream>>>(ei, deg, E);

    // ---- layer 1: (1 -> 36) ----
    {
        constexpr int DIN = 1, DOUT = 36, DD = DIN * DOUT;
        (void)hipMemsetAsync(agg, 0, (size_t)Nn * DOUT * sizeof(float), stream);
        size_t shmem = (size_t)(7 * DD + WAVES * 16 * (DIN + DOUT)) * sizeof(float);
        nnconv_edge_kernel<DIN, DOUT><<<E / (WAVES * 16), 256, shmem, stream>>>(
            x, eattr, ei, w_nn[0], b_nn[0], agg, E);
        node_update_kernel<DIN, DOUT><<<(Nn * DOUT + 255) / 256, 256, 0, stream>>>(
            agg, deg, x, lin[0], bias[0], bufA, Nn);
    }
    // ---- layer 2: (36 -> 24) ----
    {
        constexpr int DIN = 36, DOUT = 24, DD = DIN * DOUT;
        (void)hipMemsetAsync(agg, 0, (size_t)Nn * DOUT * sizeof(float), stream);
        size_t shmem = (size_t)(7 * DD + WAVES * 16 * (DIN + DOUT)) * sizeof(float);
        nnconv_edge_kernel<DIN, DOUT><<<E / (WAVES * 16), 256, shmem, stream>>>(
            bufA, eattr, ei, w_nn[1], b_nn[1], agg, E);
        node_update_kernel<DIN, DOUT><<<(Nn * DOUT + 255) / 256, 256, 0, stream>>>(
            agg, deg, bufA, lin[1], bias[1], bufB, Nn);
    }
    // ---- layer 3: (24 -> 5) ----
    {
        constexpr int DIN = 24, DOUT = 5, DD = DIN * DOUT;
        (void)hipMemsetAsync(agg, 0, (size_t)Nn * DOUT * sizeof(float), stream);
        size_t shmem = (size_t)(7 * DD + WAVES * 16 * (DIN + DOUT)) * sizeof(float);
        nnconv_edge_kernel<DIN, DOUT><<<E / (WAVES * 16), 256, shmem, stream>>>(
            bufB, eattr, ei, w_nn[2], b_nn[2], agg, E);
        node_update_kernel<DIN, DOUT><<<(Nn * DOUT + 255) / 256, 256, 0, stream>>>(
            agg, deg, bufB, lin[2], bias[2], bufA, Nn);
    }

    // final pairwise reduction: grid 4 x 4096, each thread one v4f
    cbt_kernel<<<dim3(Nn / 1024, Nn), 256, 0, stream>>>(bufA, (float*)d_out, Nn);
}